// SignalDualBackground_81930796138715
// MI455X (gfx1250) — compile-verified
//
#include <hip/hip_runtime.h>

typedef float v2f __attribute__((ext_vector_type(2)));
typedef float v8f __attribute__((ext_vector_type(8)));

typedef int v4i __attribute__((vector_size(4 * sizeof(int))));
typedef __attribute__((address_space(1))) v4i gbl_v4i;   // __device__ (global)
typedef __attribute__((address_space(3))) v4i lds_v4i;   // __shared__ (LDS)

#define B_  16
#define C_  128
#define S_  64
#define T_  1024
#define TC  128                 // time-chunk per LDS stage
#define NCHUNK (T_ / TC)        // 8
#define NTHREADS 256            // 8 wave32s; 8 x 16-column WMMA groups = 128 cols
#define NSTAGE ((S_ * TC / 4) / NTHREADS)   // 8 float4 transfers per thread

#if __has_builtin(__builtin_amdgcn_global_load_async_to_lds_b128)
#define USE_ASYNC_LDS 1
#else
#define USE_ASYNC_LDS 0
#endif

static __device__ __forceinline__ void wait_asynccnt0() {
#if __has_builtin(__builtin_amdgcn_s_wait_asynccnt)
    __builtin_amdgcn_s_wait_asynccnt(0);
#else
    asm volatile("s_wait_asynccnt 0x0" ::: "memory");
#endif
}

// One block per (b,c) series. Stream 64xTC tiles through LDS:
//   1) global -> LDS via GLOBAL_LOAD_ASYNC_TO_LDS_B128 (ASYNCcnt, 8 in flight)
//   2) synapse mean via V_WMMA_F32_16X16X4_F32 with A = 1/64 ones
//   3) EMA scan of the chunk on one wave (shfl-based linear-recurrence scan)
//   4) out = (1 - spikes) * stat from the LDS tile (b128 stores)
// spikes are touched in HBM exactly once each way -> ~1 GB total traffic (~44us floor).
__global__ __launch_bounds__(NTHREADS)
void SignalDualBackground_kernel(const float* __restrict__ spikes,
                                 const float* __restrict__ betap,
                                 float* __restrict__ out)
{
    __shared__ __align__(16) float tile[S_][TC];   // 32 KB
    __shared__ __align__(16) float xbuf[TC];
    __shared__ __align__(16) float statbuf[TC];
    __shared__ float carryLDS;

    const int tid  = threadIdx.x;
    const int lane = tid & 31;
    const int wave = tid >> 5;
    const int bc   = blockIdx.x;                   // 0 .. B_*C_-1

    const float beta   = betap[0];
    const float ombeta = 1.0f - beta;

    const float* gbase = spikes + (size_t)bc * S_ * T_;
    float*       obase = out    + (size_t)bc * S_ * T_;

    if (tid == 0) carryLDS = 0.0f;

    // A-matrix (16x4 f32): all entries 1/64 -> D[m][n] = (1/64) * sum_k B[k][n]
    v2f a; a.x = 1.0f / 64.0f; a.y = 1.0f / 64.0f;

    for (int ch = 0; ch < NCHUNK; ++ch) {
        const int t0 = ch * TC;

        __syncthreads();  // tile/statbuf from previous chunk fully consumed

        // ---- stage chunk: 64 x 128 floats, 2048 float4s, 8 per thread ----
#if USE_ASYNC_LDS
        #pragma unroll
        for (int it = 0; it < NSTAGE; ++it) {
            int idx = it * NTHREADS + tid;        // float4 index in tile
            int s   = idx >> 5;                   // row (TC/4 == 32 float4/row)
            int c4  = idx & 31;
            __builtin_amdgcn_global_load_async_to_lds_b128(
                (gbl_v4i*)(void*)(gbase + s * T_ + t0 + c4 * 4),
                (lds_v4i*)(void*)&tile[s][c4 * 4],
                /*offset=*/0, /*cpol=*/0);
        }
        wait_asynccnt0();
#else
        {
            float4 v[NSTAGE];
            #pragma unroll
            for (int it = 0; it < NSTAGE; ++it) {
                int idx = it * NTHREADS + tid;
                int s   = idx >> 5;
                int c4  = idx & 31;
                v[it] = *(const float4*)(gbase + s * T_ + t0 + c4 * 4);
            }
            #pragma unroll
            for (int it = 0; it < NSTAGE; ++it) {
                int idx = it * NTHREADS + tid;
                int s   = idx >> 5;
                int c4  = idx & 31;
                *(float4*)&tile[s][c4 * 4] = v[it];
            }
        }
#endif
        __syncthreads();

        // ---- synapse mean via WMMA: each wave owns 16 time columns ----
        {
            const int half = lane >> 4;           // 0: K rows 0/1, 1: K rows 2/3
            const int n    = lane & 15;           // column within group
            const int tg   = wave * 16;           // this wave's column base

            // preload all 16 B operands, then run the WMMA chain back-to-back
            v2f bv[16];
            #pragma unroll
            for (int i = 0; i < 16; ++i) {
                const int s0 = i * 4;
                bv[i].x = tile[s0 + 2 * half    ][tg + n];
                bv[i].y = tile[s0 + 2 * half + 1][tg + n];
            }
            v8f c = {0.f, 0.f, 0.f, 0.f, 0.f, 0.f, 0.f, 0.f};
            #pragma unroll
            for (int i = 0; i < 16; ++i) {
                c = __builtin_amdgcn_wmma_f32_16x16x4_f32(
                        false, a, false, bv[i], (short)0, c, false, false);
            }
            // D rows are identical; row M=0 lives in c[0] on lanes 0..15 (N=lane)
            if (lane < 16) xbuf[tg + lane] = c[0];
        }
        __syncthreads();

        // ---- EMA scan of 128 chunk values on wave 0 ----
        if (tid < 32) {
            float carry = carryLDS;
            float4 xe = *(const float4*)&xbuf[lane * 4];

            // local transform of this lane's 4 elements: T(s) = p*s + u
            float u = ombeta * xe.x;            float p = beta;
            u = fmaf(beta, u, ombeta * xe.y);   p *= beta;
            u = fmaf(beta, u, ombeta * xe.z);   p *= beta;
            u = fmaf(beta, u, ombeta * xe.w);   p *= beta;

            // inclusive scan of transforms across lanes: self ∘ left
            #pragma unroll
            for (int off = 1; off < 32; off <<= 1) {
                float pl = __shfl_up(p, (unsigned)off, 32);
                float ul = __shfl_up(u, (unsigned)off, 32);
                if (lane >= off) { u = fmaf(p, ul, u); p *= pl; }
            }
            // exclusive prefix (state entering this lane)
            float pe = __shfl_up(p, 1u, 32);
            float ue = __shfl_up(u, 1u, 32);
            if (lane == 0) { pe = 1.0f; ue = 0.0f; }
            float s = fmaf(pe, carry, ue);

            float s0v = fmaf(beta, s,   ombeta * xe.x);
            float s1v = fmaf(beta, s0v, ombeta * xe.y);
            float s2v = fmaf(beta, s1v, ombeta * xe.z);
            float s3v = fmaf(beta, s2v, ombeta * xe.w);
            *(float4*)&statbuf[lane * 4] = make_float4(s0v, s1v, s2v, s3v);
            if (lane == 31) carryLDS = s3v;
        }
        __syncthreads();

        // ---- emit: out = (1 - spikes) * stat, from the LDS tile ----
        #pragma unroll
        for (int it = 0; it < NSTAGE; ++it) {
            int idx = it * NTHREADS + tid;
            int s   = idx >> 5;
            int c4  = idx & 31;
            float4 sp = *(const float4*)&tile[s][c4 * 4];
            float4 st = *(const float4*)&statbuf[c4 * 4];
            float4 o;
            o.x = (1.0f - sp.x) * st.x;
            o.y = (1.0f - sp.y) * st.y;
            o.z = (1.0f - sp.z) * st.z;
            o.w = (1.0f - sp.w) * st.w;
            *(float4*)(obase + s * T_ + t0 + c4 * 4) = o;
        }
    }
}

extern "C" void kernel_launch(void* const* d_in, const int* in_sizes, int n_in,
                              void* d_out, int out_size, void* d_ws, size_t ws_size,
                              hipStream_t stream) {
    (void)in_sizes; (void)n_in; (void)out_size; (void)d_ws; (void)ws_size;
    const float* spikes = (const float*)d_in[0];
    const float* beta   = (const float*)d_in[1];
    float*       out    = (float*)d_out;

    dim3 grid(B_ * C_);       // 2048 blocks, one per (batch, channel)
    dim3 block(NTHREADS);
    SignalDualBackground_kernel<<<grid, block, 0, stream>>>(spikes, beta, out);
}